// custom_GRU_TL_23527830847612
// MI455X (gfx1250) — compile-verified
//
#include <hip/hip_runtime.h>
#include <hip/hip_bf16.h>
#include <math.h>

// ---------------------------------------------------------------------------
// TCL3D GRU for MI455X / gfx1250 (wave32, WMMA f32 16x16x4).
//  Phase 1: grid = T*B, 256 thr. standardize x -> per-gate affine -> three
//           mode-GEMMs (V_WMMA_F32_16X16X4_F32, LDS-staged permutes) ->
//           rank-1 bias -> LN -> Wx in d_ws.  Memory bound (~180MB @ 23.3TB/s).
//  Phase 2: grid = B, 128 thr (4 waves). Serial scan; waves 0..2 = gates,
//           wave 3 prefetches Wx[t]. Recurrent weights preloaded: A-fragments
//           live in registers for all 512 steps; LN affines + rank-1 bias
//           cached in LDS.
// Zero-padded weight tiles are staged in LDS so A-fragment gathers are plain
// ds_load_b32 (no predicated global loads / EXEC save-restore in inner loops).
// ---------------------------------------------------------------------------

typedef float v2f __attribute__((ext_vector_type(2)));
typedef float v8f __attribute__((ext_vector_type(8)));

#define LN_EPS 1e-5f

// D = A(16x4) * B(4x16) + C, f32 (8-arg probe-confirmed form).
__device__ __forceinline__ v8f wmma4(v2f a, v2f b, v8f c) {
  return __builtin_amdgcn_wmma_f32_16x16x4_f32(false, a, false, b, (short)0, c,
                                               false, false);
}

// Fragment layouts (wave32):
//  A 16x4 f32 : lane L -> M = L&15 ; VGPR{0,1} -> K = 4*kk + 2*(L>>4) + {0,1}
//  B 4x16 f32 : lane L -> N = L&15 ; VGPR{0,1} -> K = 4*kk + 2*(L>>4) + {0,1}
//  C/D 16x16  : lane L, VGPR v -> M = v + 8*(L>>4), N = L&15

__device__ __forceinline__ void block_reduce2(float& s, float& ss,
                                              float* scratch, int nwaves) {
#pragma unroll
  for (int off = 16; off > 0; off >>= 1) {
    s  += __shfl_xor(s, off, 32);
    ss += __shfl_xor(ss, off, 32);
  }
  int wid = threadIdx.x >> 5, lid = threadIdx.x & 31;
  if (lid == 0) { scratch[wid] = s; scratch[wid + 8] = ss; }
  __syncthreads();
  if (wid == 0) {
    float a = (lid < nwaves) ? scratch[lid] : 0.f;
    float b = (lid < nwaves) ? scratch[lid + 8] : 0.f;
#pragma unroll
    for (int off = 16; off > 0; off >>= 1) {
      a += __shfl_xor(a, off, 32);
      b += __shfl_xor(b, off, 32);
    }
    if (lid == 0) { scratch[0] = a; scratch[8] = b; }
  }
  __syncthreads();
  s = scratch[0];
  ss = scratch[8];
  __syncthreads();
}

// ===========================================================================
// Phase 1
// ===========================================================================
__global__ void __launch_bounds__(256)
tgru_phase1(const float* __restrict__ x,
            const float* __restrict__ W_fa, const float* __restrict__ W_fb,
            const float* __restrict__ W_fc,
            const float* __restrict__ W_ba, const float* __restrict__ W_bb,
            const float* __restrict__ W_bc,
            const float* __restrict__ lninw, const float* __restrict__ lninb,
            const float* __restrict__ lnoutw, const float* __restrict__ lnoutb,
            float* __restrict__ wx) {
  __shared__ __align__(16) float ls_xh[4096];  // standardized x [a][b][c]
  __shared__ float ls_xg[4096];                // per-gate affine
  __shared__ float ls_t1[16 * 256];            // [d(16)][b*16+c]
  __shared__ float ls_t2[16 * 128];            // [e(16)][d*16+c]
  __shared__ float ls_t3[16 * 64];             // [f(16)][e*8+d]
  __shared__ float ls_w[3 * 3 * 256];          // zero-padded Wf{a,b,c}[g] 16x16
  __shared__ float red[16];

  const int t = blockIdx.x >> 4;
  const int b = blockIdx.x & 15;
  const int tid = threadIdx.x;
  const int wave = tid >> 5, lane = tid & 31;
  const int m = lane & 15, kh = lane >> 4;

  // ---- stage zero-padded weight tiles: ls_w[mat*768 + g*256 + k*16 + m] ---
  for (int i = tid; i < 768; i += 256) {
    const int m_ = i & 15, k_ = (i >> 4) & 15, g_ = i >> 8;
    const int src = g_ * 128 + k_ * 8 + m_;
    ls_w[i]        = (m_ < 8) ? W_fa[src] : 0.f;
    ls_w[768 + i]  = (m_ < 8) ? W_fb[src] : 0.f;
    ls_w[1536 + i] = (m_ < 8) ? W_fc[src] : 0.f;
  }

  // ---- load (b128) + standardize x[t,b] over 4096 feats -------------------
  const float4* xp4 =
      reinterpret_cast<const float4*>(x + ((size_t)t * 16 + b) * 4096);
  float s = 0.f, ss = 0.f;
  for (int i = tid; i < 1024; i += 256) {
    float4 v = xp4[i];
    reinterpret_cast<float4*>(ls_xh)[i] = v;
    s += v.x + v.y + v.z + v.w;
    ss += v.x * v.x + v.y * v.y + v.z * v.z + v.w * v.w;
  }
  block_reduce2(s, ss, red, 8);
  const float mu = s * (1.f / 4096.f);
  const float rs = rsqrtf(ss * (1.f / 4096.f) - mu * mu + LN_EPS);
  for (int i = tid; i < 4096; i += 256) ls_xh[i] = (ls_xh[i] - mu) * rs;
  __syncthreads();

  for (int g = 0; g < 3; ++g) {
    // ---- per-gate input LN affine ----------------------------------------
    const float* lw = lninw + g * 4096;
    const float* lb = lninb + g * 4096;
    for (int i = tid; i < 4096; i += 256) ls_xg[i] = ls_xh[i] * lw[i] + lb[i];

    // ---- hoisted A-fragments for all 3 stages of this gate (ds loads) ----
    v2f aA[4], aB[4], aC[4];
#pragma unroll
    for (int kk = 0; kk < 4; ++kk) {
      const int k0 = kk * 4 + 2 * kh;
      aA[kk].x = ls_w[g * 256 + (k0 + 0) * 16 + m];
      aA[kk].y = ls_w[g * 256 + (k0 + 1) * 16 + m];
      aB[kk].x = ls_w[768 + g * 256 + (k0 + 0) * 16 + m];
      aB[kk].y = ls_w[768 + g * 256 + (k0 + 1) * 16 + m];
      aC[kk].x = ls_w[1536 + g * 256 + (k0 + 0) * 16 + m];
      aC[kk].y = ls_w[1536 + g * 256 + (k0 + 1) * 16 + m];
    }
    __syncthreads();

    // ---- Stage A: t1[d, b*16+c] = sum_a Wfa[a,d] xg[a, b*16+c] -----------
    for (int tile = wave * 2; tile < wave * 2 + 2; ++tile) {
      const int n = tile * 16 + m;
      v8f acc = {0.f, 0.f, 0.f, 0.f, 0.f, 0.f, 0.f, 0.f};
#pragma unroll
      for (int kk = 0; kk < 4; ++kk) {
        const int k0 = kk * 4 + 2 * kh;
        v2f bf;
        bf.x = ls_xg[(k0 + 0) * 256 + n];
        bf.y = ls_xg[(k0 + 1) * 256 + n];
        acc = wmma4(aA[kk], bf, acc);
      }
#pragma unroll
      for (int v = 0; v < 8; ++v) ls_t1[(v + 8 * kh) * 256 + n] = acc[v];
    }
    __syncthreads();

    // ---- Stage B: t2[e, d*16+c] = sum_b Wfb[b,e] t1[d, b*16+c] -----------
    {
      const int j = wave * 16 + m;  // d*16 + c
      const int dd = j >> 4, cc = j & 15;
      v8f acc = {0.f, 0.f, 0.f, 0.f, 0.f, 0.f, 0.f, 0.f};
#pragma unroll
      for (int kk = 0; kk < 4; ++kk) {
        const int k0 = kk * 4 + 2 * kh;
        v2f bf;
        bf.x = ls_t1[dd * 256 + (k0 + 0) * 16 + cc];
        bf.y = ls_t1[dd * 256 + (k0 + 1) * 16 + cc];
        acc = wmma4(aB[kk], bf, acc);
      }
#pragma unroll
      for (int v = 0; v < 8; ++v) ls_t2[(v + 8 * kh) * 128 + j] = acc[v];
    }
    __syncthreads();

    // ---- Stage C: t3[f, e*8+d] = sum_c Wfc[c,f] t2[e, d*16+c] ------------
    if (wave < 4) {
      const int j = wave * 16 + m;  // e*8 + d
      const int ee = j >> 3, dd = j & 7;
      v8f acc = {0.f, 0.f, 0.f, 0.f, 0.f, 0.f, 0.f, 0.f};
#pragma unroll
      for (int kk = 0; kk < 4; ++kk) {
        const int k0 = kk * 4 + 2 * kh;
        v2f bf;
        bf.x = ls_t2[ee * 128 + dd * 16 + (k0 + 0)];
        bf.y = ls_t2[ee * 128 + dd * 16 + (k0 + 1)];
        acc = wmma4(aC[kk], bf, acc);
      }
#pragma unroll
      for (int v = 0; v < 8; ++v) ls_t3[(v + 8 * kh) * 64 + j] = acc[v];
    }
    __syncthreads();

    // ---- rank-1 bias + LayerNorm(512) + affine -> Wx ---------------------
    float s2 = 0.f, ss2 = 0.f, uu[2];
#pragma unroll
    for (int r = 0; r < 2; ++r) {
      const int i = tid + r * 256;  // i = (d*8+e)*8+f
      const int dd = i >> 6, ee = (i >> 3) & 7, ff = i & 7;
      float u = ls_t3[ff * 64 + ee * 8 + dd] +
                W_ba[g * 8 + dd] * W_bb[g * 8 + ee] * W_bc[g * 8 + ff];
      uu[r] = u;
      s2 += u;
      ss2 += u * u;
    }
    block_reduce2(s2, ss2, red, 8);
    const float mu2 = s2 * (1.f / 512.f);
    const float rs2 = rsqrtf(ss2 * (1.f / 512.f) - mu2 * mu2 + LN_EPS);
#pragma unroll
    for (int r = 0; r < 2; ++r) {
      const int i = tid + r * 256;
      float o = (uu[r] - mu2) * rs2 * lnoutw[g * 512 + i] + lnoutb[g * 512 + i];
      wx[(((size_t)t * 3 + g) * 16 + b) * 512 + i] = o;
    }
    __syncthreads();
  }
}

// ===========================================================================
// Phase 2
// ===========================================================================
__global__ void __launch_bounds__(128)
tgru_phase2(const float* __restrict__ h0,
            const float* __restrict__ U_fa, const float* __restrict__ U_fb,
            const float* __restrict__ U_fc,
            const float* __restrict__ U_ba, const float* __restrict__ U_bb,
            const float* __restrict__ U_bc,
            const float* __restrict__ lninw, const float* __restrict__ lninb,
            const float* __restrict__ lnoutw, const float* __restrict__ lnoutb,
            const float* __restrict__ wx, float* __restrict__ out) {
  __shared__ float h[512];
  __shared__ float hstd[512];
  __shared__ float hg[3][512];
  __shared__ float t1s[3][16 * 64];  // [p][e*8+f]
  __shared__ float t2s[3][16 * 64];  // [q][p*8+f]
  __shared__ float t3s[3][16 * 64];  // [r][q*8+p]
  __shared__ float uh[3][512];
  __shared__ __align__(16) float wxb[3][512];
  __shared__ float ls_u[3 * 3 * 128];  // padded Uf{a,b,c}[g] 8x16
  __shared__ float ls_liw[3 * 512], ls_lib[3 * 512];
  __shared__ float ls_low[3 * 512], ls_lob[3 * 512];
  __shared__ float ls_ubias[3 * 512];
  __shared__ float red[16];

  const int b = blockIdx.x;
  const int tid = threadIdx.x;
  const int wave = tid >> 5, lane = tid & 31;
  const int m = lane & 15, kh = lane >> 4;

  // ---- one-time staging: h0, padded U tiles, LN affines, rank-1 bias -----
  for (int i = tid; i < 512; i += 128) h[i] = h0[(size_t)b * 512 + i];
  for (int i = tid; i < 384; i += 128) {
    const int m_ = i & 15, k_ = (i >> 4) & 7, g_ = i >> 7;
    const int src = g_ * 64 + k_ * 8 + m_;
    ls_u[i]       = (m_ < 8) ? U_fa[src] : 0.f;
    ls_u[384 + i] = (m_ < 8) ? U_fb[src] : 0.f;
    ls_u[768 + i] = (m_ < 8) ? U_fc[src] : 0.f;
  }
  for (int i = tid; i < 1536; i += 128) {
    ls_liw[i] = lninw[i];
    ls_lib[i] = lninb[i];
    ls_low[i] = lnoutw[i];
    ls_lob[i] = lnoutb[i];
    const int g_ = i >> 9, j = i & 511;
    const int pp = j >> 6, qq = (j >> 3) & 7, rr = j & 7;
    ls_ubias[i] = U_ba[g_ * 8 + pp] * U_bb[g_ * 8 + qq] * U_bc[g_ * 8 + rr];
  }
  __syncthreads();

  // ---- hoist all A-fragments for the whole scan (time-invariant) ---------
  v2f aA[2], aB[2], aC[2];
  if (wave < 3) {
    const int g = wave;
#pragma unroll
    for (int kk = 0; kk < 2; ++kk) {
      const int k0 = kk * 4 + 2 * kh;
      aA[kk].x = ls_u[g * 128 + (k0 + 0) * 16 + m];
      aA[kk].y = ls_u[g * 128 + (k0 + 1) * 16 + m];
      aB[kk].x = ls_u[384 + g * 128 + (k0 + 0) * 16 + m];
      aB[kk].y = ls_u[384 + g * 128 + (k0 + 1) * 16 + m];
      aC[kk].x = ls_u[768 + g * 128 + (k0 + 0) * 16 + m];
      aC[kk].y = ls_u[768 + g * 128 + (k0 + 1) * 16 + m];
    }
  }

  for (int t = 0; t < 512; ++t) {
    // ---- standardize h ---------------------------------------------------
    float s = 0.f, ss = 0.f;
    for (int i = tid; i < 512; i += 128) {
      float v = h[i];
      s += v;
      ss += v * v;
    }
    block_reduce2(s, ss, red, 4);
    const float mu = s * (1.f / 512.f);
    const float rs = rsqrtf(ss * (1.f / 512.f) - mu * mu + LN_EPS);
    for (int i = tid; i < 512; i += 128) hstd[i] = (h[i] - mu) * rs;
    __syncthreads();

    if (wave < 3) {
      const int g = wave;  // one wave per gate; EXEC full (wave-uniform)
      for (int i = lane; i < 512; i += 32)
        hg[g][i] = hstd[i] * ls_liw[g * 512 + i] + ls_lib[g * 512 + i];

      // Stage A: t1[p, e*8+f] = sum_d Ufa[d,p] hg[d, e*8+f]
#pragma unroll
      for (int tile = 0; tile < 4; ++tile) {
        const int n = tile * 16 + m;
        v8f acc = {0.f, 0.f, 0.f, 0.f, 0.f, 0.f, 0.f, 0.f};
#pragma unroll
        for (int kk = 0; kk < 2; ++kk) {
          const int k0 = kk * 4 + 2 * kh;
          v2f bf;
          bf.x = hg[g][(k0 + 0) * 64 + n];
          bf.y = hg[g][(k0 + 1) * 64 + n];
          acc = wmma4(aA[kk], bf, acc);
        }
#pragma unroll
        for (int v = 0; v < 8; ++v) t1s[g][(v + 8 * kh) * 64 + n] = acc[v];
      }

      // Stage B: t2[q, p*8+f] = sum_e Ufb[e,q] t1[p, e*8+f]
#pragma unroll
      for (int tile = 0; tile < 4; ++tile) {
        const int j = tile * 16 + m;  // p*8 + f
        const int pp = j >> 3, ff = j & 7;
        v8f acc = {0.f, 0.f, 0.f, 0.f, 0.f, 0.f, 0.f, 0.f};
#pragma unroll
        for (int kk = 0; kk < 2; ++kk) {
          const int k0 = kk * 4 + 2 * kh;
          v2f bf;
          bf.x = t1s[g][pp * 64 + (k0 + 0) * 8 + ff];
          bf.y = t1s[g][pp * 64 + (k0 + 1) * 8 + ff];
          acc = wmma4(aB[kk], bf, acc);
        }
#pragma unroll
        for (int v = 0; v < 8; ++v) t2s[g][(v + 8 * kh) * 64 + j] = acc[v];
      }

      // Stage C: t3[r, q*8+p] = sum_f Ufc[f,r] t2[q, p*8+f]
#pragma unroll
      for (int tile = 0; tile < 4; ++tile) {
        const int j = tile * 16 + m;  // q*8 + p
        const int qq = j >> 3, pp = j & 7;
        v8f acc = {0.f, 0.f, 0.f, 0.f, 0.f, 0.f, 0.f, 0.f};
#pragma unroll
        for (int kk = 0; kk < 2; ++kk) {
          const int k0 = kk * 4 + 2 * kh;
          v2f bf;
          bf.x = t2s[g][qq * 64 + pp * 8 + (k0 + 0)];
          bf.y = t2s[g][qq * 64 + pp * 8 + (k0 + 1)];
          acc = wmma4(aC[kk], bf, acc);
        }
#pragma unroll
        for (int v = 0; v < 8; ++v) t3s[g][(v + 8 * kh) * 64 + j] = acc[v];
      }

      // rank-1 bias + wave-local LN(512) + affine -> uh[g]
      float s2 = 0.f, ss2 = 0.f, uu[16];
#pragma unroll
      for (int r = 0; r < 16; ++r) {
        const int i = lane + r * 32;  // i = p*64 + q*8 + r
        const int pp = i >> 6, qq = (i >> 3) & 7, rr = i & 7;
        float u = t3s[g][rr * 64 + qq * 8 + pp] + ls_ubias[g * 512 + i];
        uu[r] = u;
        s2 += u;
        ss2 += u * u;
      }
#pragma unroll
      for (int off = 16; off > 0; off >>= 1) {
        s2 += __shfl_xor(s2, off, 32);
        ss2 += __shfl_xor(ss2, off, 32);
      }
      const float mu2 = s2 * (1.f / 512.f);
      const float rs2 = rsqrtf(ss2 * (1.f / 512.f) - mu2 * mu2 + LN_EPS);
#pragma unroll
      for (int r = 0; r < 16; ++r) {
        const int i = lane + r * 32;
        uh[g][i] =
            (uu[r] - mu2) * rs2 * ls_low[g * 512 + i] + ls_lob[g * 512 + i];
      }
    } else {
      // wave 3: prefetch Wx[t, 0..2, b, :] (b128) under the gate GEMMs
      for (int g2 = 0; g2 < 3; ++g2) {
        const float4* src = reinterpret_cast<const float4*>(
            wx + (((size_t)t * 3 + g2) * 16 + b) * 512);
        float4* dst = reinterpret_cast<float4*>(&wxb[g2][0]);
        for (int i = lane; i < 128; i += 32) dst[i] = src[i];
      }
    }
    __syncthreads();

    // ---- GRU gating ------------------------------------------------------
    for (int i = tid; i < 512; i += 128) {
      float r = 1.f / (1.f + expf(-(wxb[0][i] + uh[0][i])));
      float z = 1.f / (1.f + expf(-(wxb[1][i] + uh[1][i])));
      float n = tanhf(wxb[2][i] + r * uh[2][i]);
      float hn = (1.f - z) * n + z * h[i];
      h[i] = hn;
      out[((size_t)t * 16 + b) * 512 + i] = hn;
    }
    __syncthreads();
  }
}

// ===========================================================================
extern "C" void kernel_launch(void* const* d_in, const int* in_sizes, int n_in,
                              void* d_out, int out_size, void* d_ws,
                              size_t ws_size, hipStream_t stream) {
  const float* x       = (const float*)d_in[0];
  const float* h0      = (const float*)d_in[1];
  const float* W_fa    = (const float*)d_in[2];
  const float* W_fb    = (const float*)d_in[3];
  const float* W_fc    = (const float*)d_in[4];
  const float* W_ba    = (const float*)d_in[5];
  const float* W_bb    = (const float*)d_in[6];
  const float* W_bc    = (const float*)d_in[7];
  const float* W_lninw = (const float*)d_in[8];
  const float* W_lninb = (const float*)d_in[9];
  const float* W_loutw = (const float*)d_in[10];
  const float* W_loutb = (const float*)d_in[11];
  const float* U_fa    = (const float*)d_in[12];
  const float* U_fb    = (const float*)d_in[13];
  const float* U_fc    = (const float*)d_in[14];
  const float* U_ba    = (const float*)d_in[15];
  const float* U_bb    = (const float*)d_in[16];
  const float* U_bc    = (const float*)d_in[17];
  const float* U_lninw = (const float*)d_in[18];
  const float* U_lninb = (const float*)d_in[19];
  const float* U_loutw = (const float*)d_in[20];
  const float* U_loutb = (const float*)d_in[21];

  float* wxbuf = (float*)d_ws;   // [512][3][16][512] = 50.3 MB
  float* out   = (float*)d_out;  // [512][16][512]

  tgru_phase1<<<dim3(512 * 16), dim3(256), 0, stream>>>(
      x, W_fa, W_fb, W_fc, W_ba, W_bb, W_bc, W_lninw, W_lninb, W_loutw,
      W_loutb, wxbuf);

  tgru_phase2<<<dim3(16), dim3(128), 0, stream>>>(
      h0, U_fa, U_fb, U_fc, U_ba, U_bb, U_bc, U_lninw, U_lninb, U_loutw,
      U_loutb, wxbuf, out);
}